// CLRNetAssign_88622355185749
// MI455X (gfx1250) — compile-verified
//
#include <hip/hip_runtime.h>
#include <hip/hip_bf16.h>
#include <math.h>

// ---------------------------------------------------------------------------
// CLRNet dynamic-k assigner for MI455X (gfx1250, wave32).
// B=32, P=1000, T=16, O=72, D=78.
// Start-point distance cross-term p·t computed with V_WMMA_F32_16X16X4_F32.
// ---------------------------------------------------------------------------

typedef float v2f __attribute__((ext_vector_type(2)));
typedef float v8f __attribute__((ext_vector_type(8)));

#define CB 32
#define CP 1000
#define CT 16
#define CO 72
#define CD 78
#define PTILE 64            // preds per workgroup (4 waves x 16)
#define NTILES ((CP + PTILE - 1) / PTILE)
#define BIG_COST 1.0e8f
#define W_REG 3.0f

// ---------------------------------------------------------------------------
// Stage 2: per-(b, 64-pred-tile) cost core.
//  - masked sum |p_off - t_off|  -> distances + line IoU (shared reduction)
//  - start-point dot product via WMMA f32 16x16x4
//  - per-batch maxima via atomicMax on float bits (all values >= 0)
// ---------------------------------------------------------------------------
__global__ __launch_bounds__(128)
void clr_cost_core(const float* __restrict__ preds,
                   const float* __restrict__ targets,
                   const int* __restrict__ imgw_p,
                   const int* __restrict__ imgh_p,
                   float* __restrict__ dist_o,
                   float* __restrict__ sdist_o,
                   float* __restrict__ liou_o,
                   unsigned* __restrict__ maxima)
{
  const int b    = blockIdx.y;
  const int tile = blockIdx.x;
  const int tid  = threadIdx.x;
  const int p0   = tile * PTILE;

  const float iw = (float)(*imgw_p);
  const float w  = iw - 1.0f;
  const float h  = (float)(*imgh_p) - 1.0f;

  __shared__ float sT[CT][CO + 1];      // scaled target offsets
  __shared__ float sP[PTILE][CO + 1];   // scaled pred offsets
  __shared__ float s_tsx[CT], s_tsy[CT], s_tn[CT], s_tcnt[CT], s_tth[CT];
  __shared__ float s_psx[PTILE], s_psy[PTILE], s_pn[PTILE], s_pth[PTILE];

  // stage target offsets (x w)
  for (int i = tid; i < CT * CO; i += 128) {
    const int t = i / CO, o = i % CO;
    sT[t][o] = targets[(size_t)(b * CT + t) * CD + 6 + o] * w;
  }
  // stage pred offsets (x w), zero-pad rows beyond P
  for (int i = tid; i < PTILE * CO; i += 128) {
    const int pl = i / CO, o = i % CO;
    const int p  = p0 + pl;
    sP[pl][o] = (p < CP) ? preds[(size_t)(b * CP + p) * CD + 6 + o] * w : 0.0f;
  }
  if (tid < CT) {
    const float* tg = &targets[(size_t)(b * CT + tid) * CD];
    const float x = tg[2] * h, y = tg[3] * w;
    s_tsx[tid] = x; s_tsy[tid] = y; s_tn[tid] = x * x + y * y; s_tth[tid] = tg[4];
  }
  if (tid < PTILE) {
    const int p = p0 + tid;
    if (p < CP) {
      const float* pd = &preds[(size_t)(b * CP + p) * CD];
      const float x = pd[2] * h, y = pd[3] * w;
      s_psx[tid] = x; s_psy[tid] = y; s_pn[tid] = x * x + y * y; s_pth[tid] = pd[4];
    } else {
      s_psx[tid] = 0.f; s_psy[tid] = 0.f; s_pn[tid] = 0.f; s_pth[tid] = 0.f;
    }
  }
  __syncthreads();

  if (tid < CT) {           // valid-offset count per target
    float c = 0.f;
    #pragma unroll 8
    for (int o = 0; o < CO; ++o) {
      const float v = sT[tid][o];
      c += (v >= 0.f && v < iw) ? 1.f : 0.f;
    }
    s_tcnt[tid] = c;
  }
  __syncthreads();

  const int lane  = tid & 31;
  const int wv    = tid >> 5;      // wave id: owns pred rows [wv*16, wv*16+16)
  const int tcol  = lane & 15;     // N index == target column (WMMA C/D layout)
  const int rhalf = lane >> 4;     // 0: M = r, 1: M = 8 + r

  const float cnt = s_tcnt[tcol];
  const float tth = s_tth[tcol];

  // masked sum |p - t| ; layout matches WMMA accumulator: row M = rhalf*8 + r
  float sumd[8];
  #pragma unroll
  for (int r = 0; r < 8; ++r) sumd[r] = 0.f;

  const float* tr = sT[tcol];
  for (int o = 0; o < CO; ++o) {
    const float tv    = tr[o];
    const bool  valid = (tv >= 0.f && tv < iw);
    #pragma unroll
    for (int r = 0; r < 8; ++r) {
      const int pl = wv * 16 + rhalf * 8 + r;
      const float d = fabsf(sP[pl][o] - tv);
      sumd[r] += valid ? d : 0.f;
    }
  }

  // ---- WMMA: D[p][t] = psx[p]*tsx[t] + psy[p]*tsy[t] (K=4, upper K zero) ----
  // A 16x4 f32: lanes 0-15 carry {K0,K1} for M=lane; lanes 16-31 carry {K2,K3}=0
  // B 4x16 f32: lanes 0-15 carry rows {K0,K1} for N=lane; lanes 16-31 zero
  v2f av, bv;
  if (lane < 16) {
    av.x = s_psx[wv * 16 + lane]; av.y = s_psy[wv * 16 + lane];
    bv.x = s_tsx[lane];           bv.y = s_tsy[lane];
  } else {
    av.x = 0.f; av.y = 0.f; bv.x = 0.f; bv.y = 0.f;
  }
  v8f acc = {};
  acc = __builtin_amdgcn_wmma_f32_16x16x4_f32(
      /*neg_a=*/false, av, /*neg_b=*/false, bv,
      /*c_mod=*/(short)0, acc, /*reuse_a=*/false, /*reuse_b=*/false);

  float mD = 0.f, mS = 0.f, mT = 0.f;
  #pragma unroll
  for (int r = 0; r < 8; ++r) {
    const int pl = wv * 16 + rhalf * 8 + r;
    const int p  = p0 + pl;
    const float sd_sum = sumd[r];
    const float d   = sd_sum / fmaxf(cnt, 1.f);
    const float io  = (30.f * cnt - sd_sum) / (30.f * cnt + sd_sum + 1e-9f);
    const float sq  = s_pn[pl] + s_tn[tcol] - 2.f * acc[r];
    const float sdv = sqrtf(fmaxf(sq, 0.f));
    const float tdv = fabsf(s_pth[pl] - tth) * 180.f;
    if (p < CP) {
      const size_t idx = (size_t)(b * CP + p) * CT + tcol;
      dist_o[idx]  = d;
      sdist_o[idx] = sdv;
      liou_o[idx]  = io;
      mD = fmaxf(mD, d); mS = fmaxf(mS, sdv); mT = fmaxf(mT, tdv);
    }
  }
  atomicMax(&maxima[b * 3 + 0], __float_as_uint(mD));
  atomicMax(&maxima[b * 3 + 1], __float_as_uint(mS));
  atomicMax(&maxima[b * 3 + 2], __float_as_uint(mT));
}

// ---------------------------------------------------------------------------
// Stage 3: normalize scores with batch maxima, add focal cost, apply mask.
// One thread per (b,p), loops over T=16.
// ---------------------------------------------------------------------------
__global__ void clr_total_cost(const float* __restrict__ preds,
                               const float* __restrict__ targets,
                               const int* __restrict__ masks,
                               const float* __restrict__ dist_i,
                               const float* __restrict__ sdist_i,
                               const unsigned* __restrict__ maxima,
                               float* __restrict__ cost_o)
{
  const int gid = blockIdx.x * blockDim.x + threadIdx.x;
  if (gid >= CB * CP) return;
  const int b = gid / CP;

  const float dmax = fmaxf(__uint_as_float(maxima[b * 3 + 0]), 1e-8f);
  const float smax = fmaxf(__uint_as_float(maxima[b * 3 + 1]), 1e-8f);
  const float tmax = fmaxf(__uint_as_float(maxima[b * 3 + 2]), 1e-8f);

  const float* pd = &preds[(size_t)gid * CD];
  // focal cost for both class hypotheses (alpha=0.25, gamma=2)
  float cls[2];
  #pragma unroll
  for (int c = 0; c < 2; ++c) {
    const float x  = pd[c];
    const float pr = 1.f / (1.f + __expf(-x));
    const float pos = -__logf(pr + 1e-12f) * 0.25f * (1.f - pr) * (1.f - pr);
    const float neg = -__logf(1.f - pr + 1e-12f) * 0.75f * pr * pr;
    cls[c] = pos - neg;
  }
  const float pth = pd[4];

  #pragma unroll
  for (int t = 0; t < CT; ++t) {
    const size_t idx = (size_t)gid * CT + t;
    const float* tg = &targets[(size_t)(b * CT + t) * CD];
    const float ds = 1.f - dist_i[idx] / dmax + 0.01f;
    const float ss = 1.f - sdist_i[idx] / smax + 0.01f;
    const float td = fabsf(pth - tg[4]) * 180.f;
    const float ts = 1.f - td / tmax + 0.01f;
    const float reg = fmaxf(ds, 0.001f) * fmaxf(ss, 0.001f) * fmaxf(ts, 0.001f);
    int lab = (int)tg[1];
    lab = (lab != 0) ? 1 : 0;
    float tc = -reg * reg * W_REG + cls[lab];
    if (masks[b * CT + t] <= 0) tc = BIG_COST;
    cost_o[idx] = tc;
  }
}

// ---------------------------------------------------------------------------
// Stage 4: one wave per (b,t). dynamic_k = clamp(int(sum top4 IoU),1,4);
// then mark the dynamic_k lowest-cost preds in the matching matrix.
// Tie-break: lower pred index wins (matches jax.lax.top_k / argmin).
// ---------------------------------------------------------------------------
__global__ __launch_bounds__(32)
void clr_topk(const float* __restrict__ liou_i,
              const float* __restrict__ cost_i,
              const int* __restrict__ masks,
              float* __restrict__ matching)
{
  const int bt = blockIdx.x;
  if (masks[bt] <= 0) return;
  const int b = bt / CT, t = bt % CT;
  const int lane = threadIdx.x;

  // ---- top-4 IoU values -> dynamic k ----
  int selI[4];
  float sum = 0.f;
  for (int j = 0; j < 4; ++j) {
    float bestV = -1e30f; int bestI = 0x7fffffff;
    for (int p = lane; p < CP; p += 32) {
      bool skip = false;
      for (int k = 0; k < j; ++k) skip |= (p == selI[k]);
      if (skip) continue;
      const float v = liou_i[(size_t)(b * CP + p) * CT + t];
      if (v > bestV || (v == bestV && p < bestI)) { bestV = v; bestI = p; }
    }
    for (int off = 16; off > 0; off >>= 1) {
      const float ov = __shfl_xor(bestV, off, 32);
      const int   oi = __shfl_xor(bestI, off, 32);
      if (ov > bestV || (ov == bestV && oi < bestI)) { bestV = ov; bestI = oi; }
    }
    selI[j] = bestI;
    sum += bestV;
  }
  int dyn = (int)sum;                 // truncation toward zero, as astype(int32)
  if (dyn < 1) dyn = 1;
  if (dyn > 4) dyn = 4;

  // ---- dynamic_k lowest-cost preds for this target ----
  int chosen[4];
  for (int j = 0; j < dyn; ++j) {
    float bestV = 1e38f; int bestI = 0x7fffffff;
    for (int p = lane; p < CP; p += 32) {
      bool skip = false;
      for (int k = 0; k < j; ++k) skip |= (p == chosen[k]);
      if (skip) continue;
      const float v = cost_i[(size_t)(b * CP + p) * CT + t];
      if (v < bestV || (v == bestV && p < bestI)) { bestV = v; bestI = p; }
    }
    for (int off = 16; off > 0; off >>= 1) {
      const float ov = __shfl_xor(bestV, off, 32);
      const int   oi = __shfl_xor(bestI, off, 32);
      if (ov < bestV || (ov == bestV && oi < bestI)) { bestV = ov; bestI = oi; }
    }
    chosen[j] = bestI;
    if (lane == 0)
      matching[(size_t)(b * CP + bestI) * CT + t] = 1.0f;
  }
}

// ---------------------------------------------------------------------------
// Stage 5: per-(b,p) conflict resolution + outputs.
// out[0 .. B*P)      : has_match as 1.0/0.0
// out[B*P .. 2*B*P)  : matched target index (or -1) as float
// ---------------------------------------------------------------------------
__global__ void clr_resolve(const float* __restrict__ matching,
                            const float* __restrict__ cost_i,
                            float* __restrict__ out)
{
  const int gid = blockIdx.x * blockDim.x + threadIdx.x;
  if (gid >= CB * CP) return;
  const float* mrow = &matching[(size_t)gid * CT];
  const float* crow = &cost_i[(size_t)gid * CT];

  int count = 0, single = -1;
  #pragma unroll
  for (int t = 0; t < CT; ++t) {
    if (mrow[t] > 0.5f) { ++count; if (single < 0) single = t; }
  }
  int matched;
  if (count == 0) {
    matched = -1;
  } else if (count == 1) {
    matched = single;
  } else {  // conflict: argmin cost among matched targets (first-min wins)
    float best = INFINITY; int bi = 0;
    #pragma unroll
    for (int t = 0; t < CT; ++t) {
      if (mrow[t] > 0.5f && crow[t] < best) { best = crow[t]; bi = t; }
    }
    matched = bi;
  }
  out[gid]           = (matched >= 0) ? 1.0f : 0.0f;
  out[CB * CP + gid] = (float)matched;
}

// ---------------------------------------------------------------------------
extern "C" void kernel_launch(void* const* d_in, const int* in_sizes, int n_in,
                              void* d_out, int out_size, void* d_ws, size_t ws_size,
                              hipStream_t stream)
{
  const float* preds   = (const float*)d_in[0];
  const float* targets = (const float*)d_in[1];
  const int*   masks   = (const int*)d_in[2];
  const int*   imgw    = (const int*)d_in[3];
  const int*   imgh    = (const int*)d_in[4];

  const size_t NPT = (size_t)CB * CP * CT;   // 512,000
  float* ws       = (float*)d_ws;
  float* dist     = ws;
  float* sdist    = dist  + NPT;
  float* liou     = sdist + NPT;
  float* cost     = liou  + NPT;
  float* matching = cost  + NPT;
  unsigned* maxima = (unsigned*)(matching + NPT);

  // zero matching matrix + per-batch maxima (contiguous region)
  hipMemsetAsync(matching, 0, (NPT + (size_t)CB * 3) * sizeof(float), stream);

  clr_cost_core<<<dim3(NTILES, CB), 128, 0, stream>>>(
      preds, targets, imgw, imgh, dist, sdist, liou, maxima);

  clr_total_cost<<<(CB * CP + 255) / 256, 256, 0, stream>>>(
      preds, targets, masks, dist, sdist, maxima, cost);

  clr_topk<<<CB * CT, 32, 0, stream>>>(liou, cost, masks, matching);

  clr_resolve<<<(CB * CP + 255) / 256, 256, 0, stream>>>(
      matching, cost, (float*)d_out);
}